// EmlMasterFormula_3891240370257
// MI455X (gfx1250) — compile-verified
//
#include <hip/hip_runtime.h>
#include <stdint.h>

#define EXP_CLAMP 500.0f
#define VAL_CLAMP 1.0e6f
#define PI_F      3.14159265358979323846f

#define TPB 256      // threads per block
#define BPB 64       // batch elements per block (4 quadrant threads each)
#define CW  32       // DFS iterations per leaf-value chunk (64 leaf rows / quadrant)

typedef float v2f __attribute__((ext_vector_type(2)));
typedef float v8f __attribute__((ext_vector_type(8)));

// ---- gfx1250 feature guards ----
#if defined(__has_builtin)
#if __has_builtin(__builtin_amdgcn_global_load_async_to_lds_b32) && \
    __has_builtin(__builtin_amdgcn_s_wait_asynccnt)
#define CDNA5_ASYNC 1
#endif
#if __has_builtin(__builtin_amdgcn_wmma_f32_16x16x4_f32)
#define CDNA5_WMMA 1
#endif
#endif

#ifdef CDNA5_ASYNC
__device__ __forceinline__ void async_g2l_b32(const void* g, void* l) {
    // generic->AS1 keeps the 64-bit address; generic LDS pointer's low 32 bits
    // are the LDS offset, so truncation yields a valid AS3 pointer.
    __builtin_amdgcn_global_load_async_to_lds_b32(
        (__attribute__((address_space(1))) int*)(uintptr_t)g,
        (__attribute__((address_space(3))) int*)(uint32_t)(uintptr_t)l,
        0, 0);
}
#endif

// ---- helpers ----
__device__ __forceinline__ float scrub1(float v) {
    v = (v == v) ? v : 0.0f;                        // NaN -> 0
    return fminf(fmaxf(v, -VAL_CLAMP), VAL_CLAMP);  // +-inf -> +-1e6, clamp
}

__device__ __forceinline__ float fast_atan2f(float y, float x) {
    float ax = __builtin_fabsf(x), ay = __builtin_fabsf(y);
    float mx = fmaxf(ax, ay), mn = fminf(ax, ay);
    float a  = (mx == 0.0f) ? 0.0f : mn * __builtin_amdgcn_rcpf(mx);
    float s  = a * a;
    float p  = -0.01172120f;
    p = fmaf(s, p,  0.05265332f);
    p = fmaf(s, p, -0.11643287f);
    p = fmaf(s, p,  0.19354346f);
    p = fmaf(s, p, -0.33262347f);
    p = fmaf(s, p,  0.99997726f);
    float r = a * p;
    r = (ay > ax)  ? (1.57079637f - r) : r;
    r = (x < 0.0f) ? (PI_F - r) : r;
    return copysignf(r, y);
}

// scrub(safe_exp(a) - conj(log(conj(b)))) for complex a=(are,aim), b=(bre,bim)
__device__ __forceinline__ float2 eml_scrub(float are, float aim, float bre, float bim) {
    float r  = fminf(fmaxf(are, -EXP_CLAMP), EXP_CLAMP);
    float e  = __expf(r);
    float Er = e * __cosf(aim);
    float Ei = e * __sinf(aim);
    float m2 = fmaf(bre, bre, bim * bim);
    float Lr = 0.5f * __logf(m2);
    float Li = -fast_atan2f(-bim, bre);             // conj(log(conj(b))).imag
    float2 o;
    o.x = scrub1(Er - Lr);
    o.y = scrub1(Ei - Li);
    return o;
}

// ---- main tree kernel: thread = (batch element, level-2 quadrant) ----
__global__ __launch_bounds__(TPB) void eml_tree_kernel(
    const float* __restrict__ x,
    const float* __restrict__ leaf_logits,   // (1024,2)
    const float* __restrict__ int_logits,    // (1022,3)
    float* __restrict__ out,                 // pred interleaved (re,im) at [0,32768)
    int batch)
{
    __shared__ float2 sLeaf[1024];   // softmax leaf weights (w0,w1)
    __shared__ float4 sInt[1022];    // softmax internal weights (w0,w1,w2,pad)
    // lv: leaf-value chunk, layout idx = cc*256 + b_local*4 + q (conflict-free reads)
    // also aliased as the raw-logit staging buffer (disjoint lifetimes)
    __shared__ float  lv[64 * BPB * 4];      // 16384 floats = 64KB
    __shared__ float  xs[BPB];               // block's x values
    __shared__ float2 sComb[TPB];            // quadrant results

    float* sRaw = lv;                        // staging alias (5114 <= 16384 floats)
    const int tid = threadIdx.x;

    // 1) stage raw logits to LDS (gfx1250 async-to-LDS path)
#ifdef CDNA5_ASYNC
    for (int k = tid; k < 2048; k += TPB) async_g2l_b32(leaf_logits + k, &sRaw[k]);
    for (int k = tid; k < 3066; k += TPB) async_g2l_b32(int_logits + k, &sRaw[2048 + k]);
    __builtin_amdgcn_s_wait_asynccnt(0);
#else
    for (int k = tid; k < 2048; k += TPB) sRaw[k]        = leaf_logits[k];
    for (int k = tid; k < 3066; k += TPB) sRaw[2048 + k] = int_logits[k];
#endif
    if (tid < BPB) {
        int gb = blockIdx.x * BPB + tid;
        xs[tid] = (gb < batch) ? x[gb] : 0.0f;
    }
    __syncthreads();

    // 2) softmax weights into LDS
    for (int r = tid; r < 1024; r += TPB) {
        float a = sRaw[2*r], b = sRaw[2*r + 1];
        float m = fmaxf(a, b);
        float e0 = __expf(a - m), e1 = __expf(b - m);
        float inv = __builtin_amdgcn_rcpf(e0 + e1);
        sLeaf[r] = make_float2(e0 * inv, e1 * inv);
    }
    for (int r = tid; r < 1022; r += TPB) {
        float a = sRaw[2048 + 3*r], b = sRaw[2048 + 3*r + 1], c = sRaw[2048 + 3*r + 2];
        float m = fmaxf(fmaxf(a, b), c);
        float e0 = __expf(a - m), e1 = __expf(b - m), e2 = __expf(c - m);
        float inv = __builtin_amdgcn_rcpf(e0 + e1 + e2);
        sInt[r] = make_float4(e0 * inv, e1 * inv, e2 * inv, 0.0f);
    }

    const int q    = tid & 3;                   // level-2 subtree quadrant
    const int bl   = tid >> 2;                  // batch element within block
    const int b    = blockIdx.x * BPB + bl;
    const bool valid = (b < batch);
    const float xv = valid ? x[b] : 0.0f;
    const int wave = tid >> 5;                  // 8 wave32 waves per block
    const int lane = tid & 31;

    // post-order weight-row base for subtree of node (2,q): {0,255,511,766}
    int wi = q * 255 + (q >> 1);

    // iterative post-order fold; top of stack in registers, spill array <=7 deep
    float arr_re[8], arr_im[8];
    int   sp  = 0;
    float cre = 0.0f, cim = 0.0f;
    const int ib = bl * 4 + q;                  // this thread's slot in lv rows

    for (int chunk = 0; chunk < 128 / CW; ++chunk) {
        __syncthreads();   // previous chunk fully consumed / weights ready

        // --- leaf-value stage: lv[cc][b_local][quadrant] for 64 rows per quadrant ---
#ifdef CDNA5_WMMA
        // 64 tiles of 16(leaf-rows) x 16(batch) per chunk; 8 tiles per wave.
        // A(16x4 f32): lanes 0-15 -> K=0,1 (=w0,w1); lanes 16-31 -> K=2,3 (=0).
        // B(4x16 f32): lanes 0-15 -> K=0 (=1.0) / K=1 (=x); lanes 16-31 -> K=2,3 (=0).
        for (int i = 0; i < 8; ++i) {
            int t  = wave * 8 + i;              // tile id 0..63
            int tq = t >> 4;                    // quadrant of this tile
            int tr = (t >> 2) & 3;              // row-tile (16 rows) within 64-row range
            int tb = t & 3;                     // batch-tile (16 batches)
            int m  = lane & 15;
            bool lo = (lane < 16);
            float2 wrow = sLeaf[tq * 256 + chunk * 64 + tr * 16 + m];
            v2f A, B;
            A.x = lo ? wrow.x : 0.0f;           // K=0 -> w0
            A.y = lo ? wrow.y : 0.0f;           // K=1 -> w1
            B.x = lo ? 1.0f : 0.0f;             // K=0 row: ones
            B.y = lo ? xs[tb * 16 + m] : 0.0f;  // K=1 row: x
            v8f acc = {};
            acc = __builtin_amdgcn_wmma_f32_16x16x4_f32(
                false, A, false, B, (short)0, acc, false, false);
            int bb = tb * 16 + m;
            int r0 = tr * 16 + (lo ? 0 : 8);
#pragma unroll
            for (int j = 0; j < 8; ++j)
                lv[(r0 + j) * 256 + bb * 4 + tq] = acc[j];
        }
#else
        // scalar fallback: each thread fills its own slots
        for (int cc = 0; cc < 64; ++cc) {
            float2 wv = sLeaf[q * 256 + chunk * 64 + cc];
            lv[cc * 256 + ib] = fmaf(wv.y, xv, wv.x);
        }
#endif
        __syncthreads();

        // --- DFS over this chunk's CW iterations ---
        for (int ci = 0; ci < CW; ++ci) {
            int c = chunk * CW + ci;
            float la = lv[(2 * ci)     * 256 + ib];
            float lb = lv[(2 * ci + 1) * 256 + ib];

            // bottom eml, specialized for real inputs
            float r   = fminf(fmaxf(la, -EXP_CLAMP), EXP_CLAMP);
            float e   = __expf(r);
            float Lr  = __logf(__builtin_fabsf(lb));
            float vre = scrub1(e - Lr);
            float vim = (lb < 0.0f) ? -PI_F : 0.0f;   // - conj(log(conj(b))).imag

            // affine transform (post-order weight row), push
            float4 w = sInt[wi]; ++wi;
            float tre = fmaf(w.y, xv, w.x) + w.z * vre;
            float tim = w.z * vim;
            if (c != 0) { arr_re[sp] = cre; arr_im[sp] = cim; ++sp; }
            cre = tre; cim = tim;

            // merge cascade: ctz(c+1) merges (uniform across block -> no divergence)
            int nm = __builtin_ctz(c + 1);
            for (int k = 0; k < nm; ++k) {
                --sp;
                float2 v = eml_scrub(arr_re[sp], arr_im[sp], cre, cim);
                float4 w2 = sInt[wi]; ++wi;
                cre = fmaf(w2.y, xv, w2.x) + w2.z * v.x;
                cim = w2.z * v.y;
            }
        }
    }

    // (cre,cim) is the transformed level-2 node for this quadrant
    sComb[tid] = make_float2(cre, cim);
    __syncthreads();

    if (q == 0 && valid) {
        float2 t0 = sComb[tid], t1 = sComb[tid + 1];
        float2 t2 = sComb[tid + 2], t3 = sComb[tid + 3];
        // level-1 left node: weight row widx(1,0)=510
        float2 vL = eml_scrub(t0.x, t0.y, t1.x, t1.y);
        float4 wL = sInt[510];
        float TLre = fmaf(wL.y, xv, wL.x) + wL.z * vL.x;
        float TLim = wL.z * vL.y;
        // level-1 right node: weight row widx(1,1)=1021
        float2 vR = eml_scrub(t2.x, t2.y, t3.x, t3.y);
        float4 wR = sInt[1021];
        float TRre = fmaf(wR.y, xv, wR.x) + wR.z * vR.x;
        float TRim = wR.z * vR.y;
        // root (no transform)
        float2 root = eml_scrub(TLre, TLim, TRre, TRim);
        ((float2*)out)[b] = root;
    }
}

// ---- small softmax outputs: leaf_probs (1024x2) and int_probs (1022x3) ----
__global__ __launch_bounds__(256) void probs_kernel(
    const float* __restrict__ leaf_logits,
    const float* __restrict__ int_logits,
    float* __restrict__ out)
{
    int i = blockIdx.x * blockDim.x + threadIdx.x;
    if (i < 1024) {
        float a = leaf_logits[2*i], b = leaf_logits[2*i + 1];
        float m = fmaxf(a, b);
        float e0 = __expf(a - m), e1 = __expf(b - m);
        float inv = __builtin_amdgcn_rcpf(e0 + e1);
        out[32768 + 2*i]     = e0 * inv;
        out[32768 + 2*i + 1] = e1 * inv;
    } else if (i < 2046) {
        int j = i - 1024;
        float a = int_logits[3*j], b = int_logits[3*j + 1], c = int_logits[3*j + 2];
        float m = fmaxf(fmaxf(a, b), c);
        float e0 = __expf(a - m), e1 = __expf(b - m), e2 = __expf(c - m);
        float inv = __builtin_amdgcn_rcpf(e0 + e1 + e2);
        out[34816 + 3*j]     = e0 * inv;
        out[34816 + 3*j + 1] = e1 * inv;
        out[34816 + 3*j + 2] = e2 * inv;
    }
}

extern "C" void kernel_launch(void* const* d_in, const int* in_sizes, int n_in,
                              void* d_out, int out_size, void* d_ws, size_t ws_size,
                              hipStream_t stream) {
    const float* x    = (const float*)d_in[0];
    const float* leaf = (const float*)d_in[1];
    const float* inte = (const float*)d_in[2];
    float* out = (float*)d_out;
    const int batch  = in_sizes[0];                 // 16384
    const int blocks = (batch + BPB - 1) / BPB;     // 256
    eml_tree_kernel<<<blocks, TPB, 0, stream>>>(x, leaf, inte, out, batch);
    probs_kernel<<<8, 256, 0, stream>>>(leaf, inte, out);
}